// DGCNN_66735201845496
// MI455X (gfx1250) — compile-verified
//
#include <hip/hip_runtime.h>

// ---------------------------------------------------------------------------
// DGCNN forward for MI455X (gfx1250). fp32 WMMA (v_wmma_f32_16x16x4_f32)
// for all GEMMs: Gram/distance matrices, gathered edge-conv GEMMs (two-pass
// recompute around batchnorm), and the dense head. wave32 tiles, 4 waves/block.
// ---------------------------------------------------------------------------

typedef float v2f __attribute__((ext_vector_type(2)));
typedef float v8f __attribute__((ext_vector_type(8)));

#define NBATCH 8
#define NP     2048
#define KNN    10
#define SLOTS  64
#define EPSBN  1e-5f
#define NEGINF (-1e30f)

__device__ __forceinline__ v8f wmma_f32(v2f a, v2f b, v8f c) {
  // D = A(16x4) * B(4x16) + C(16x16), fp32, wave32
  return __builtin_amdgcn_wmma_f32_16x16x4_f32(false, a, false, b,
                                               (short)0, c, false, false);
}

__device__ __forceinline__ void atomicMaxFloat(float* addr, float val) {
  // IEEE-ordered float max via int/uint atomics (works for mixed signs)
  if (val >= 0.0f) atomicMax((int*)addr, __float_as_int(val));
  else             atomicMin((unsigned int*)addr, __float_as_uint(val));
}

// ---------------------------------------------------------------- utilities
__global__ void fill_kernel(float* __restrict__ p, float v, size_t n) {
  size_t i = (size_t)blockIdx.x * blockDim.x + threadIdx.x;
  if (i < n) p[i] = v;
}

// x (S=2048, B=8, K=8, D=3) -> pts (B, N=2048, C=24)
__global__ void transpose_kernel(const float* __restrict__ x,
                                 float* __restrict__ pts) {
  int i = blockIdx.x * blockDim.x + threadIdx.x;
  const int total = NP * NBATCH * 24;
  if (i >= total) return;
  int c = i % 24;
  int t = i / 24;
  int s = t % NP;
  int b = t / NP;
  pts[i] = x[(((size_t)s * NBATCH + b) * 8 + (c / 3)) * 3 + (c % 3)];
}

// squared norms per point: xx[b*NP+n]
__global__ void sqnorm_kernel(const float* __restrict__ X, int ld, int Cin,
                              float* __restrict__ xx) {
  int i = blockIdx.x * blockDim.x + threadIdx.x;
  if (i >= NBATCH * NP) return;
  const float* p = X + (size_t)i * ld;
  float s = 0.f;
  for (int c = 0; c < Cin; ++c) s += p[c] * p[c];
  xx[i] = s;
}

// ------------------------------------------------- pairwise distance (Gram)
// D[m][n] = 2*dot(x_m, x_n) - xx[m] - xx[n], one batch; one wave per 16x16 tile.
__global__ __launch_bounds__(128)
void dist_kernel(const float* __restrict__ X, int ld, int Cin,
                 const float* __restrict__ xx, float* __restrict__ D, int b) {
  int wave = threadIdx.x >> 5, lane = threadIdx.x & 31;
  int tm = (blockIdx.x * 4 + wave) * 16;
  int tn = blockIdx.y * 16;
  int half = lane >> 4, r = lane & 15;
  const float* Xb = X + (size_t)b * NP * ld;
  const float* arow = Xb + (size_t)(tm + r) * ld;
  const float* brow = Xb + (size_t)(tn + r) * ld;
  v8f acc = {};
  for (int k0 = 0; k0 < Cin; k0 += 4) {
    int c0 = k0 + 2 * half;
    v2f a, bf;
    a[0] = arow[c0];  a[1] = arow[c0 + 1];
    bf[0] = brow[c0]; bf[1] = brow[c0 + 1];
    acc = wmma_f32(a, bf, acc);
  }
  const float* xxb = xx + b * NP;
  int ncol = tn + r;
  float xn = xxb[ncol];
  for (int i = 0; i < 8; ++i) {
    int m = tm + i + 8 * half;
    D[(size_t)m * NP + ncol] = 2.f * acc[i] - xxb[m] - xn;
  }
}

// ----------------------------------------------------------------- top-k(10)
// One wave per row; per-lane sorted top-10 over a strided slice, LDS merge.
// Tie-break matches jax.lax.top_k: equal values -> smaller index first.
__global__ __launch_bounds__(32)
void topk_kernel(const float* __restrict__ D, int* __restrict__ idx, int b) {
  int n = blockIdx.x;
  int lane = threadIdx.x;
  const float* row = D + (size_t)n * NP;
  float v[KNN]; int vi[KNN];
  for (int i = 0; i < KNN; ++i) { v[i] = NEGINF; vi[i] = 0x7fffffff; }
  for (int m = lane; m < NP; m += 32) {
    float d = row[m];
    if (d > v[KNN - 1] || (d == v[KNN - 1] && m < vi[KNN - 1])) {
      int p = KNN - 1;
      while (p > 0 && (d > v[p - 1] || (d == v[p - 1] && m < vi[p - 1]))) {
        v[p] = v[p - 1]; vi[p] = vi[p - 1]; --p;
      }
      v[p] = d; vi[p] = m;
    }
  }
  __shared__ float sv[32 * KNN];
  __shared__ int   si[32 * KNN];
  for (int i = 0; i < KNN; ++i) { sv[lane * KNN + i] = v[i]; si[lane * KNN + i] = vi[i]; }
  __syncthreads();
  if (lane == 0) {
    int head[32];
    for (int t = 0; t < 32; ++t) head[t] = 0;
    int* out = idx + ((size_t)b * NP + n) * KNN;
    for (int j = 0; j < KNN; ++j) {
      int bt = 0; float bv = NEGINF; int bi = 0x7fffffff;
      for (int t = 0; t < 32; ++t) {
        if (head[t] < KNN) {
          float cv = sv[t * KNN + head[t]];
          int   ci = si[t * KNN + head[t]];
          if (cv > bv || (cv == bv && ci < bi)) { bv = cv; bi = ci; bt = t; }
        }
      }
      out[j] = bi;
      head[bt]++;
    }
  }
}

// -------------------------------------------- edge-conv GEMM (two passes)
// Rows m = ((b*NP + n)*KNN + j), A[m][c] built on the fly from gathered
// neighbors (no feat materialization). pass 0: accumulate BN sums (slotted
// atomics). pass 1: recompute, apply BN scale/shift + leaky, fold max over j
// into Xout via float atomic-max.
__global__ __launch_bounds__(128)
void edgeconv_kernel(const float* __restrict__ X, int ld, int Cin,
                     const int* __restrict__ idx,
                     const float* __restrict__ W, int O,
                     float* __restrict__ sumS, float* __restrict__ sqS,
                     const float* __restrict__ bnSS,
                     float* __restrict__ Xout, int ldo, int chanOff,
                     int pass) {
  const int twoC = 2 * Cin;
  const int ROWS_PER_B = NP * KNN;
  int wave = threadIdx.x >> 5, lane = threadIdx.x & 31;
  int tileId = blockIdx.x * 4 + wave;
  int tileM = tileId * 16;
  int tileN = blockIdx.y * 16;
  int half = lane >> 4, r = lane & 15;

  int rowA = tileM + r;
  int b   = rowA / ROWS_PER_B;
  int mib = rowA - b * ROWS_PER_B;
  int n   = mib / KNN;
  int j   = mib - n * KNN;
  int nb  = idx[(size_t)(b * NP + n) * KNN + j];

  const float* Xb  = X + (size_t)b * NP * ld;
  const float* xn  = Xb + (size_t)n  * ld;
  const float* xnb = Xb + (size_t)nb * ld;
  const float* wr  = W + (size_t)(tileN + r) * twoC;

  v8f acc = {};
  for (int k0 = 0; k0 < twoC; k0 += 4) {
    int c0 = k0 + 2 * half;
    int c1 = c0 + 1;
    v2f a, bf;
    a[0] = (c0 < Cin) ? (xnb[c0] - xn[c0]) : xn[c0 - Cin];
    a[1] = (c1 < Cin) ? (xnb[c1] - xn[c1]) : xn[c1 - Cin];
    bf[0] = wr[c0];
    bf[1] = wr[c1];
    acc = wmma_f32(a, bf, acc);
  }

  int col = tileN + r;
  if (pass == 0) {
    float s = 0.f, q = 0.f;
    for (int i = 0; i < 8; ++i) { float hv = acc[i]; s += hv; q += hv * hv; }
    s += __shfl_down(s, 16, 32);
    q += __shfl_down(q, 16, 32);
    if (half == 0) {
      int slot = tileId & (SLOTS - 1);
      atomicAdd(&sumS[col * SLOTS + slot], s);
      atomicAdd(&sqS [col * SLOTS + slot], q);
    }
  } else {
    float sc = bnSS[col], sh = bnSS[O + col];
    float* outB = Xout + (size_t)b * NP * ldo + chanOff + col;
    int curn = -1; float cmax = NEGINF;
    for (int i = 0; i < 8; ++i) {
      int m  = tileM + i + 8 * half;          // 8 consecutive rows
      int nn = (m - b * ROWS_PER_B) / KNN;
      float hv = acc[i] * sc + sh;
      hv = hv > 0.f ? hv : 0.2f * hv;          // leaky relu
      if (nn != curn) {
        if (curn >= 0) atomicMaxFloat(outB + (size_t)curn * ldo, cmax);
        curn = nn; cmax = hv;
      } else {
        cmax = fmaxf(cmax, hv);
      }
    }
    atomicMaxFloat(outB + (size_t)curn * ldo, cmax);
  }
}

// Reduce slotted sums -> BN scale/shift: y = h*scale + shift
__global__ void bn_finalize_kernel(const float* __restrict__ sumS,
                                   const float* __restrict__ sqS,
                                   const float* __restrict__ gamma,
                                   const float* __restrict__ beta,
                                   float* __restrict__ bnSS, int O, float cnt) {
  int o = blockIdx.x * blockDim.x + threadIdx.x;
  if (o >= O) return;
  float s = 0.f, q = 0.f;
  for (int t = 0; t < SLOTS; ++t) { s += sumS[o * SLOTS + t]; q += sqS[o * SLOTS + t]; }
  float mu  = s / cnt;
  float var = q / cnt - mu * mu;
  float sc  = gamma[o] * rsqrtf(var + EPSBN);
  bnSS[o]     = sc;
  bnSS[O + o] = beta[o] - mu * sc;
}

// ------------------------------------------------------------- dense head
// C[m][o] = act(A[m][:] . W[o][:] + bias[o]); O may be ragged (552, 255).
__global__ __launch_bounds__(128)
void dense_kernel(const float* __restrict__ A, int lda, int Ksz,
                  const float* __restrict__ W, int O,
                  const float* __restrict__ bias,
                  float* __restrict__ Cout, int ldc, int act) {
  int wave = threadIdx.x >> 5, lane = threadIdx.x & 31;
  int tm = (blockIdx.x * 4 + wave) * 16;
  int tn = blockIdx.y * 16;
  int half = lane >> 4, r = lane & 15;
  bool colOK = (tn + r) < O;
  const float* arow = A + (size_t)(tm + r) * lda;
  const float* wr   = W + (size_t)(colOK ? (tn + r) : 0) * Ksz;
  v8f acc = {};
  for (int k0 = 0; k0 < Ksz; k0 += 4) {
    int c0 = k0 + 2 * half;
    v2f a, bf;
    a[0] = arow[c0]; a[1] = arow[c0 + 1];
    bf[0] = colOK ? wr[c0] : 0.f;
    bf[1] = colOK ? wr[c0 + 1] : 0.f;
    acc = wmma_f32(a, bf, acc);
  }
  int col = tn + r;
  if (col < O) {
    float bv = bias ? bias[col] : 0.f;
    for (int i = 0; i < 8; ++i) {
      int m = tm + i + 8 * half;
      float v = acc[i] + bv;
      if (act) v = v > 0.f ? v : 0.2f * v;
      Cout[(size_t)m * ldc + col] = v;
    }
  }
}

// ---------------------------------------------------------------------------
extern "C" void kernel_launch(void* const* d_in, const int* in_sizes, int n_in,
                              void* d_out, int out_size, void* d_ws, size_t ws_size,
                              hipStream_t stream) {
  (void)in_sizes; (void)n_in; (void)out_size; (void)ws_size;
  const float* x   = (const float*)d_in[0];
  const float* W1  = (const float*)d_in[1];
  const float* g1  = (const float*)d_in[2];
  const float* b1  = (const float*)d_in[3];
  const float* W2  = (const float*)d_in[4];
  const float* g2  = (const float*)d_in[5];
  const float* b2  = (const float*)d_in[6];
  const float* W3  = (const float*)d_in[7];
  const float* g3  = (const float*)d_in[8];
  const float* b3  = (const float*)d_in[9];
  const float* W4  = (const float*)d_in[10];
  const float* g4  = (const float*)d_in[11];
  const float* b4  = (const float*)d_in[12];
  const float* W5  = (const float*)d_in[13];
  const float* W6  = (const float*)d_in[14];
  const float* Wd1 = (const float*)d_in[15];
  const float* bd1 = (const float*)d_in[16];
  const float* Wd2 = (const float*)d_in[17];
  const float* bd2 = (const float*)d_in[18];

  const size_t BN = (size_t)NBATCH * NP;
  float* ws = (float*)d_ws;
  size_t off = 0;
  float* pts  = ws + off; off += BN * 24;     // 1.6 MB
  float* cat  = ws + off; off += BN * 512;    // 32 MB  (x1|x2|x3|x4 slices)
  float* h5   = ws + off; off += BN * 256;    // 16 MB
  float* tmp  = ws + off; off += BN * 552;    // 36 MB
  float* xx   = ws + off; off += BN;          // sq norms
  float* dist = ws + off; off += (size_t)NP * NP;  // 16.8 MB, one batch
  int*   idx  = (int*)(ws + off); off += BN * KNN;
  float* sumS = ws + off; off += 256 * SLOTS;
  float* sqS  = ws + off; off += 256 * SLOTS;
  float* bnSS = ws + off; off += 512;

  float* feature = (float*)d_out;                 // (B,N,552)
  float* logits  = feature + BN * 552;            // (B,N,255)

  // init accumulator buffer (atomic-max target) every call -> deterministic
  {
    size_t n = BN * 512;
    fill_kernel<<<dim3((unsigned)((n + 255) / 256)), 256, 0, stream>>>(cat, NEGINF, n);
  }
  transpose_kernel<<<dim3((NP * NBATCH * 24 + 255) / 256), 256, 0, stream>>>(x, pts);

  struct Layer {
    const float* X; int ld, C;
    const float* W; int O;
    const float* g; const float* be;
    int chanOff;
  } layers[4] = {
    { pts,       24,  24,  W1,  64, g1, b1,   0 },
    { cat,       512, 64,  W2,  64, g2, b2,  64 },
    { cat + 64,  512, 64,  W3, 128, g3, b3, 128 },
    { cat + 128, 512, 128, W4, 256, g4, b4, 256 },
  };

  for (int L = 0; L < 4; ++L) {
    const Layer& ly = layers[L];
    sqnorm_kernel<<<dim3((NBATCH * NP + 255) / 256), 256, 0, stream>>>(ly.X, ly.ld, ly.C, xx);
    for (int b = 0; b < NBATCH; ++b) {
      dist_kernel<<<dim3(NP / 64, NP / 16), 128, 0, stream>>>(ly.X, ly.ld, ly.C, xx, dist, b);
      topk_kernel<<<dim3(NP), 32, 0, stream>>>(dist, idx, b);
    }
    fill_kernel<<<dim3((2 * 256 * SLOTS + 255) / 256), 256, 0, stream>>>(sumS, 0.f, 2 * 256 * SLOTS);
    dim3 eg((unsigned)(BN * KNN / 16 / 4), (unsigned)(ly.O / 16));
    edgeconv_kernel<<<eg, 128, 0, stream>>>(ly.X, ly.ld, ly.C, idx, ly.W, ly.O,
                                            sumS, sqS, bnSS, cat, 512, ly.chanOff, 0);
    bn_finalize_kernel<<<dim3((ly.O + 63) / 64), 64, 0, stream>>>(
        sumS, sqS, ly.g, ly.be, bnSS, ly.O, (float)(BN * KNN));
    edgeconv_kernel<<<eg, 128, 0, stream>>>(ly.X, ly.ld, ly.C, idx, ly.W, ly.O,
                                            sumS, sqS, bnSS, cat, 512, ly.chanOff, 1);
  }

  // dense head: cat(512) -> h5(256) -> feature(552) -> tmp(552) -> logits(255)
  dim3 dgrid((unsigned)(BN / 64), 0, 1);
  dgrid.y = 256 / 16;
  dense_kernel<<<dgrid, 128, 0, stream>>>(cat, 512, 512, W5, 256, nullptr, h5, 256, 1);
  dgrid.y = (552 + 15) / 16;
  dense_kernel<<<dgrid, 128, 0, stream>>>(h5, 256, 256, W6, 552, nullptr, feature, 552, 1);
  dense_kernel<<<dgrid, 128, 0, stream>>>(feature, 552, 552, Wd1, 552, bd1, tmp, 552, 0);
  dgrid.y = (255 + 15) / 16;
  dense_kernel<<<dgrid, 128, 0, stream>>>(tmp, 552, 552, Wd2, 255, bd2, logits, 255, 0);
}